// HeteGAT_multi_59150289601067
// MI455X (gfx1250) — compile-verified
//
#include <hip/hip_runtime.h>

// Shapes from the reference
#define P_  3
#define N_  6144
#define B_  3072
#define F_  512
#define H_  8
#define O_  32
#define D_  256   // H*O
#define A_  128
#define HI_ 64    // F/H

typedef float v2f __attribute__((ext_vector_type(2)));
typedef float v8f __attribute__((ext_vector_type(8)));

// ---------------------------------------------------------------------------
// Kernel 1: feature gather + per-head projection (seq) + attention scalars f1/f2
// grid: P*H*(B/8), block 256 (8 waves); wave -> one b row, lane -> o (0..31)
// ---------------------------------------------------------------------------
__global__ __launch_bounds__(256)
void proj_kernel(const float* __restrict__ features, const int* __restrict__ batch_nodes,
                 const float* __restrict__ Wc1, const float* __restrict__ bc1,
                 const float* __restrict__ a1, const float* __restrict__ ab1,
                 const float* __restrict__ a2, const float* __restrict__ ab2,
                 float* __restrict__ seq, float* __restrict__ f1, float* __restrict__ f2)
{
    const int bt   = blockIdx.x % (B_ / 8);
    const int h    = (blockIdx.x / (B_ / 8)) % H_;
    const int p    = blockIdx.x / ((B_ / 8) * H_);
    const int lane = threadIdx.x & 31;
    const int wv   = threadIdx.x >> 5;
    const int b    = bt * 8 + wv;

    __shared__ float sW[HI_ * O_];   // sW[i*32 + o]
    __shared__ float sX[8][HI_];

    // Wc1[p][h][o][i] -> LDS transposed for conflict-free lane access
    const float* Wbase = Wc1 + ((size_t)(p * H_ + h)) * O_ * HI_;
    for (int t = threadIdx.x; t < O_ * HI_; t += 256) {
        int o = t / HI_, i = t % HI_;
        sW[i * O_ + o] = Wbase[t];
    }
    const int idx = batch_nodes[p * B_ + b];
    const float* frow = features + ((size_t)p * N_ + (size_t)idx) * F_ + h * HI_;
    sX[wv][lane]      = frow[lane];
    sX[wv][lane + 32] = frow[lane + 32];
    __syncthreads();

    const int o = lane;
    float acc = 0.f;
#pragma unroll
    for (int i = 0; i < HI_; ++i) acc = fmaf(sX[wv][i], sW[i * O_ + o], acc);
    acc += bc1[(p * H_ + h) * O_ + o];

    seq[(((size_t)(p * H_ + h)) * B_ + b) * O_ + o] = acc;

    // wave reduction over o for f1/f2
    float p1 = acc * a1[(p * H_ + h) * O_ + o];
    float p2 = acc * a2[(p * H_ + h) * O_ + o];
#pragma unroll
    for (int s = 16; s > 0; s >>= 1) {
        p1 += __shfl_xor(p1, s, 32);
        p2 += __shfl_xor(p2, s, 32);
    }
    if (lane == 0) {
        f1[(p * H_ + h) * B_ + b] = p1 + ab1[p * H_ + h];
        f2[(p * H_ + h) * B_ + b] = p2 + ab2[p * H_ + h];
    }
}

// ---------------------------------------------------------------------------
// Kernel 2: flash-style attention. One wave per (p, h, 16-row q tile).
// Streams k in chunks of 4: p-values placed in the f32 16x4 A layout,
// seq slices in the 4x16 B layout, accumulated with v_wmma_f32_16x16x4_f32.
// ---------------------------------------------------------------------------
__global__ __launch_bounds__(32)
void attn_kernel(const float* __restrict__ biases, const int* __restrict__ batch_nodes,
                 const float* __restrict__ seq, const float* __restrict__ f1,
                 const float* __restrict__ f2, float* __restrict__ h1)
{
    const int QT  = B_ / 16;
    const int qt  = blockIdx.x % QT;
    const int h   = (blockIdx.x / QT) % H_;
    const int p   = blockIdx.x / (QT * H_);

    const int lane = threadIdx.x;
    const int m    = lane & 15;      // A: row q; B: col o
    const int grp  = lane >> 4;      // selects K pair {0,1} vs {2,3}

    const int q0 = qt * 16;
    const int qA = q0 + m;
    const int* idx = batch_nodes + p * B_;
    const float* bbrow = biases + (size_t)p * N_ * N_ + (size_t)idx[qA] * N_;

    const float* f1H  = f1 + (p * H_ + h) * B_;
    const float* f2H  = f2 + (p * H_ + h) * B_;
    const float* seqH = seq + ((size_t)(p * H_ + h)) * B_ * O_;
    const float  f1q  = f1H[qA];

    v8f acc_lo = {};
    v8f acc_hi = {};
    float psum = 0.f;

    for (int kb = 0; kb < B_; kb += 4) {
        const int k0 = kb + grp * 2;
        const int k1 = k0 + 1;
        const int c0 = idx[k0];
        const int c1 = idx[k1];

        float l0 = f1q + f2H[k0];
        float l1 = f1q + f2H[k1];
        l0 = (l0 >= 0.f) ? l0 : 0.01f * l0;     // leaky_relu(0.01)
        l1 = (l1 >= 0.f) ? l1 : 0.01f * l1;
        l0 += bbrow[c0];                         // + bias mask (double gather)
        l1 += bbrow[c1];
        const float p0 = __expf(l0);
        const float p1 = __expf(l1);
        psum += p0 + p1;

        v2f a;  a.x = p0; a.y = p1;              // A(m, k): lane = m + 16*(k>=2), vgpr = k&1

        const float* s0 = seqH + (size_t)k0 * O_;
        const float* s1 = seqH + (size_t)k1 * O_;
        v2f blo; blo.x = s0[m];      blo.y = s1[m];        // B(k, n): lane = n + 16*(k>=2)
        v2f bhi; bhi.x = s0[m + 16]; bhi.y = s1[m + 16];

        acc_lo = __builtin_amdgcn_wmma_f32_16x16x4_f32(false, a, false, blo,
                                                       (short)0, acc_lo, false, false);
        acc_hi = __builtin_amdgcn_wmma_f32_16x16x4_f32(false, a, false, bhi,
                                                       (short)0, acc_hi, false, false);
    }

    // lanes m and m+16 partition k-space for row q=m -> combine
    psum += __shfl_xor(psum, 16, 32);

    float* out = h1 + ((size_t)p * B_) * D_;
#pragma unroll
    for (int r = 0; r < 8; ++r) {
        const int   qloc = r + 8 * grp;                 // C/D row layout
        const float rs   = __shfl(psum, qloc, 32);      // rowsum lives in lane qloc
        float vlo = acc_lo[r] / rs;
        float vhi = acc_hi[r] / rs;
        vlo = (vlo > 0.f) ? vlo : (__expf(vlo) - 1.f);  // elu
        vhi = (vhi > 0.f) ? vhi : (__expf(vhi) - 1.f);
        float* orow = out + (size_t)(q0 + qloc) * D_ + h * O_;
        orow[m]      = vlo;
        orow[m + 16] = vhi;
    }
}

// ---------------------------------------------------------------------------
// Kernel 3: multi[b,p,:] = RL[p] * (h1[p][b,:] @ Wm.T + bm) via WMMA fp32.
// One wave per (p, 16-b tile, 16-d tile), K loop over D in chunks of 4.
// ---------------------------------------------------------------------------
__global__ __launch_bounds__(32)
void wmul_kernel(const float* __restrict__ h1, const float* __restrict__ Wm,
                 const float* __restrict__ bm, const float* __restrict__ RL,
                 float* __restrict__ multi)
{
    const int DT = D_ / 16;   // 16
    const int BT = B_ / 16;   // 192
    const int dt = blockIdx.x % DT;
    const int bt = (blockIdx.x / DT) % BT;
    const int p  = blockIdx.x / (DT * BT);

    const int lane = threadIdx.x;
    const int m    = lane & 15;
    const int grp  = lane >> 4;

    const int b0 = bt * 16, d0 = dt * 16;
    const float* h1p = h1 + ((size_t)p * B_) * D_;

    v8f acc = {};
    for (int e0 = 0; e0 < D_; e0 += 4) {
        const int e = e0 + grp * 2;
        v2f a;  a.x  = h1p[(size_t)(b0 + m) * D_ + e];
                a.y  = h1p[(size_t)(b0 + m) * D_ + e + 1];
        v2f b2; b2.x = Wm[(size_t)(d0 + m) * D_ + e];       // Wm^T(e, d) = Wm[d][e]
                b2.y = Wm[(size_t)(d0 + m) * D_ + e + 1];
        acc = __builtin_amdgcn_wmma_f32_16x16x4_f32(false, a, false, b2,
                                                    (short)0, acc, false, false);
    }
    const float rl = RL[p];
#pragma unroll
    for (int r = 0; r < 8; ++r) {
        const int brow = b0 + r + 8 * grp;
        const int dcol = d0 + m;
        multi[((size_t)brow * P_ + p) * D_ + dcol] = rl * (acc[r] + bm[dcol]);
    }
}

// ---------------------------------------------------------------------------
// Kernel 4: semantic attention over the 3 relations. One block per node b.
// ---------------------------------------------------------------------------
__global__ __launch_bounds__(128)
void sem_kernel(const float* __restrict__ multi, const float* __restrict__ w_omega,
                const float* __restrict__ b_omega, const float* __restrict__ u_omega,
                float* __restrict__ outp)
{
    const int b = blockIdx.x;
    __shared__ float sIn[P_ * D_];
    __shared__ float red[128];
    __shared__ float svu[P_];

    for (int t = threadIdx.x; t < P_ * D_; t += 128)
        sIn[t] = multi[(size_t)b * P_ * D_ + t];
    __syncthreads();

    const int a = threadIdx.x;   // attention dim, A_=128
    for (int p = 0; p < P_; ++p) {
        float dot = 0.f;
        for (int d = 0; d < D_; ++d)
            dot = fmaf(sIn[p * D_ + d], w_omega[(size_t)d * A_ + a], dot);
        float v = tanhf(dot + b_omega[a]);
        red[a] = v * u_omega[a];
        __syncthreads();
        for (int s = 64; s > 0; s >>= 1) {
            if (a < s) red[a] += red[a + s];
            __syncthreads();
        }
        if (a == 0) svu[p] = red[0];
        __syncthreads();
    }

    const float mx = fmaxf(svu[0], fmaxf(svu[1], svu[2]));
    const float e0 = __expf(svu[0] - mx), e1 = __expf(svu[1] - mx), e2 = __expf(svu[2] - mx);
    const float inv = 1.f / (e0 + e1 + e2);
    const float al0 = e0 * inv, al1 = e1 * inv, al2 = e2 * inv;

    for (int d = threadIdx.x; d < D_; d += 128)
        outp[(size_t)b * D_ + d] = al0 * sIn[0 * D_ + d] + al1 * sIn[1 * D_ + d] + al2 * sIn[2 * D_ + d];
}

// ---------------------------------------------------------------------------
extern "C" void kernel_launch(void* const* d_in, const int* in_sizes, int n_in,
                              void* d_out, int out_size, void* d_ws, size_t ws_size,
                              hipStream_t stream)
{
    (void)in_sizes; (void)n_in; (void)out_size; (void)ws_size;

    const float* features = (const float*)d_in[0];
    const float* biases   = (const float*)d_in[1];
    const int*   batch    = (const int*)d_in[2];
    const float* RL       = (const float*)d_in[3];
    const float* Wc1      = (const float*)d_in[4];
    const float* bc1      = (const float*)d_in[5];
    const float* a1       = (const float*)d_in[6];
    const float* ab1      = (const float*)d_in[7];
    const float* a2       = (const float*)d_in[8];
    const float* ab2      = (const float*)d_in[9];
    const float* Wm       = (const float*)d_in[10];
    const float* bm       = (const float*)d_in[11];
    const float* w_omega  = (const float*)d_in[12];
    const float* b_omega  = (const float*)d_in[13];
    const float* u_omega  = (const float*)d_in[14];
    float* out = (float*)d_out;

    // workspace layout (floats): ~28.9 MB total
    float* ws    = (float*)d_ws;
    float* seq   = ws;                                  // P*H*B*O
    float* f1    = seq + (size_t)P_ * H_ * B_ * O_;     // P*H*B
    float* f2    = f1  + (size_t)P_ * H_ * B_;          // P*H*B
    float* h1    = f2  + (size_t)P_ * H_ * B_;          // P*B*D
    float* multi = h1  + (size_t)P_ * B_ * D_;          // B*P*D

    proj_kernel<<<P_ * H_ * (B_ / 8), 256, 0, stream>>>(features, batch, Wc1, bc1,
                                                        a1, ab1, a2, ab2, seq, f1, f2);
    attn_kernel<<<P_ * H_ * (B_ / 16), 32, 0, stream>>>(biases, batch, seq, f1, f2, h1);
    wmul_kernel<<<P_ * (B_ / 16) * (D_ / 16), 32, 0, stream>>>(h1, Wm, bm, RL, multi);
    sem_kernel<<<B_, 128, 0, stream>>>(multi, w_omega, b_omega, u_omega, out);
}